// Spiking1DALIFLayer_86019605004464
// MI455X (gfx1250) — compile-verified
//
#include <hip/hip_runtime.h>

// ALIF spiking scan: B=64, C=512, T=1024. One thread owns one (b,c) chain.
// Memory-bound (~256MB traffic => ~11us floor at 23.3 TB/s). Strategy:
// async-copy x tiles into LDS (coalesced), run the serial recurrence out of
// LDS, stage spikes in LDS, async-store them back. Double-buffered with
// ASYNCcnt waits.

#define B_ 64
#define C_ 512
#define T_ 1024
#define CH 64          // chains per block (blockDim.x), 2 waves
#define TILE 32        // timesteps per tile
#define STRIDE 36      // padded row stride in floats (144B: 16B-aligned)
#define NT (T_ / TILE) // 32 tiles
#define NCHUNK (TILE / 4) // 8 float4 chunks per thread per tile

// ---------- exact builtin pointer types ----------
typedef int v4i __attribute__((ext_vector_type(4)));
typedef __attribute__((address_space(1))) v4i* g4p;
typedef __attribute__((address_space(3))) v4i* l4p;

__device__ __forceinline__ g4p to_g4(const void* p) {
  return (g4p)(unsigned long long)p;
}
__device__ __forceinline__ l4p to_l4(const void* p) {
  // flat LDS addresses: low 32 bits == LDS offset (aperture is in high bits)
  return (l4p)(unsigned)(unsigned long long)p;
}

// ---------- async copy primitives ----------
#if defined(__has_builtin)
#if __has_builtin(__builtin_amdgcn_global_load_async_to_lds_b128) && \
    __has_builtin(__builtin_amdgcn_global_store_async_from_lds_b128)
#define HAVE_ASYNC_BUILTIN 1
#endif
#if __has_builtin(__builtin_amdgcn_s_wait_asynccnt)
#define HAVE_WAIT_BUILTIN 1
#endif
#endif

__device__ __forceinline__ void async_load_f4(const float* g, float* l) {
#ifdef HAVE_ASYNC_BUILTIN
  __builtin_amdgcn_global_load_async_to_lds_b128(to_g4(g), to_l4(l), 0, 0);
#else
  unsigned lo = (unsigned)(unsigned long long)l;
  asm volatile("global_load_async_to_lds_b128 %0, %1, off"
               :: "v"(lo), "v"(g) : "memory");
#endif
}

__device__ __forceinline__ void async_store_f4(float* g, const float* l) {
#ifdef HAVE_ASYNC_BUILTIN
  __builtin_amdgcn_global_store_async_from_lds_b128(to_g4(g), to_l4(l), 0, 0);
#else
  unsigned lo = (unsigned)(unsigned long long)l;
  asm volatile("global_store_async_from_lds_b128 %0, %1, off"
               :: "v"(g), "v"(lo) : "memory");
#endif
}

#ifdef HAVE_WAIT_BUILTIN
#define WAIT_ASYNCCNT(n) __builtin_amdgcn_s_wait_asynccnt(n)
#else
#define WAIT_ASYNCCNT(n) asm volatile("s_wait_asynccnt %0" :: "n"(n) : "memory")
#endif

// ---------- tile copy helpers ----------
__device__ __forceinline__ void load_tile(const float* __restrict__ x, int n0,
                                          int t0, float* buf, int tid) {
#pragma unroll
  for (int j = 0; j < NCHUNK; ++j) {
    int cid = j * CH + tid;          // 0..511 float4 chunks in the tile
    int row = cid >> 3;              // 8 chunks (32 floats) per chain row
    int c4  = (cid & 7) * 4;         // float offset within row
    async_load_f4(x + ((size_t)(n0 + row) * T_ + t0 + c4),
                  buf + row * STRIDE + c4);
  }
}

__device__ __forceinline__ void store_tile(float* __restrict__ out, int n0,
                                           int t0, const float* buf, int tid) {
#pragma unroll
  for (int j = 0; j < NCHUNK; ++j) {
    int cid = j * CH + tid;
    int row = cid >> 3;
    int c4  = (cid & 7) * 4;
    async_store_f4(out + ((size_t)(n0 + row) * T_ + t0 + c4),
                   buf + row * STRIDE + c4);
  }
}

// ---------- kernel ----------
__global__ __launch_bounds__(CH, 1) void alif_scan_kernel(
    const float* __restrict__ x,
    const float* __restrict__ p_alpha, const float* __restrict__ p_beta,
    const float* __restrict__ p_gamma, const float* __restrict__ p_rho,
    const float* __restrict__ vth_arr, float* __restrict__ out) {
  // Double-buffered staging planes; indexed (not pointer arrays) so no
  // static addrspacecast initializers are generated.
  __shared__ float s_in[2][CH * STRIDE];
  __shared__ float s_out[2][CH * STRIDE];

  const int tid = threadIdx.x;
  const int n0 = blockIdx.x * CH;

  const float a = p_alpha[0];
  const float b = p_beta[0];
  const float g = p_gamma[0];
  const float r = p_rho[0];
  const float vth = vth_arr[(n0 + tid) & (C_ - 1)];

  // ALIF state (matches reference init: mem=inp=spk=thadapt=0, Athpot=1)
  float mem = 0.0f, inp = 0.0f, spk = 0.0f, thad = 0.0f, Ath = 1.0f;

  // prefetch tile 0
  load_tile(x, n0, 0, &s_in[0][0], tid);

  for (int k = 0; k < NT; ++k) {
    const int cur = k & 1;
    if (k + 1 < NT) {
      load_tile(x, n0, (k + 1) * TILE, &s_in[cur ^ 1][0], tid);
      // <=8 outstanding: since loads retire in order among loads and stores
      // in order among stores, this proves loads(k) done and stores(k-2)
      // done (the two buffers we are about to touch are free).
      WAIT_ASYNCCNT(8);
    } else {
      WAIT_ASYNCCNT(0);
    }
    __syncthreads();

    const float* ip = &s_in[cur][0] + tid * STRIDE;
    float* op = &s_out[cur][0] + tid * STRIDE;
#pragma unroll
    for (int t = 0; t < TILE; ++t) {
      float xv = ip[t];
      float rst = spk * Ath;            // previous spike * previous Athpot
      inp = fmaf(a, inp, xv);           // inp = alpha*inp + x_t
      mem = fmaf(b, mem, inp - rst);    // mem = beta*mem + inp - rst
      thad = fmaf(r, thad, spk);        // thadapt = rho*thadapt + spk_prev
      Ath = fmaf(g, thad, vth);         // Athpot = Vth + gamma*thadapt
      spk = (mem - Ath) > 0.0f ? 1.0f : 0.0f;
      op[t] = spk;
    }
    __syncthreads();
    store_tile(out, n0, k * TILE, &s_out[cur][0], tid);
  }
  WAIT_ASYNCCNT(0);  // drain final stores before endpgm
}

// ---------- host launch ----------
extern "C" void kernel_launch(void* const* d_in, const int* in_sizes, int n_in,
                              void* d_out, int out_size, void* d_ws,
                              size_t ws_size, hipStream_t stream) {
  (void)in_sizes; (void)n_in; (void)out_size; (void)d_ws; (void)ws_size;
  const float* x = (const float*)d_in[0];
  const float* alpha = (const float*)d_in[1];
  const float* beta = (const float*)d_in[2];
  const float* gamma = (const float*)d_in[3];
  const float* rho = (const float*)d_in[4];
  const float* vth = (const float*)d_in[5];
  float* out = (float*)d_out;

  dim3 grid((B_ * C_) / CH);  // 512 blocks
  dim3 block(CH);             // 64 threads = 2 waves
  alif_scan_kernel<<<grid, block, 0, stream>>>(x, alpha, beta, gamma, rho, vth,
                                               out);
}